// NeuralGraphCollaborativeFiltering_14843406975284
// MI455X (gfx1250) — compile-verified
//
#include <hip/hip_runtime.h>
#include <hip/hip_bf16.h>

// ---------------------------------------------------------------------------
// GNN pipeline on MI455X (gfx1250, wave32):
//   x = emb[x_idx]
//   h0 = ClusterGCN(x)  = agg(x, 1/deg) @ W_out^T + x @ W_root^T ; BN ; ReLU
//   h1 = SGConv(h0)     = gcn_norm-agg(h0) @ W_sg1^T + b_sg1     ; BN ; ReLU
//   out= SGConv(h1)     = gcn_norm-agg(h1) @ W_sg2^T + b_sg2
// Dense GEMMs use V_WMMA_F32_16X16X4_F32 (exact fp32, matches reference),
// 80-row (5 M-tile) blocking so each B fragment feeds 5 WMMAs, and A tiles
// are staged into LDS with GLOBAL_LOAD_ASYNC_TO_LDS_B128 (ASYNCcnt path).
// Sparse propagation uses one wave32 per edge + global f32 atomics (L2-resident).
// ---------------------------------------------------------------------------

#define N_FEAT 128
#define BN_EPS 1e-5f
#define ROWS_PER_BLOCK 80          // 50000 = 625 * 80; 5 full 16-row tiles
#define M_TILES 5
#define A_STRIDE 132               // 128 + 4 pad floats -> conflict-free ds_load_b64

typedef __attribute__((ext_vector_type(2))) float v2f;
typedef __attribute__((ext_vector_type(8))) float v8f;

// ---------------- utility ----------------
__global__ void k_zero(float* __restrict__ p, int n) {
    int i = blockIdx.x * blockDim.x + threadIdx.x;
    if (i < n) p[i] = 0.0f;
}

// gather rows: x[i] = emb[x_idx[i]]   (float4 per lane, coalesced)
__global__ void k_gather(const float* __restrict__ emb, const int* __restrict__ idx,
                         float* __restrict__ x, int n) {
    int i = blockIdx.x * blockDim.x + threadIdx.x;   // over n*32 float4 slots
    if (i >= n * (N_FEAT / 4)) return;
    int node = i >> 5;
    int v    = i & 31;
    const float4* src = (const float4*)(emb + (size_t)idx[node] * N_FEAT);
    float4*       dst = (float4*)(x + (size_t)node * N_FEAT);
    dst[v] = src[v];
}

// ---------------- degree / norms ----------------
__global__ void k_deg(const int* __restrict__ col, float* __restrict__ deg, int nE) {
    int e = blockIdx.x * blockDim.x + threadIdx.x;
    if (e >= nE) return;
    atomicAdd(&deg[col[e]], 1.0f);
}

__global__ void k_prep(const float* __restrict__ deg, float* __restrict__ dinv,
                       float* __restrict__ dis, int n) {
    int i = blockIdx.x * blockDim.x + threadIdx.x;
    if (i >= n) return;
    float d = deg[i];
    dinv[i] = 1.0f / fmaxf(d, 1.0f);          // ClusterGCN: 1/deg(c)
    dis[i]  = rsqrtf(d + 1.0f);               // SGConv gcn_norm: deg incl self loop
}

// ---------------- sparse propagation (one wave32 per edge) ----------------
// agg[c] += x[r] * dinv[c]
__global__ void k_cluster_scatter(const float* __restrict__ x, const int* __restrict__ row,
                                  const int* __restrict__ col, const float* __restrict__ dinv,
                                  float* __restrict__ agg, int nE) {
    int e = (blockIdx.x * blockDim.x + threadIdx.x) >> 5;
    if (e >= nE) return;
    int lane = threadIdx.x & 31;
    int r = row[e], c = col[e];
    float w = dinv[c];
    float4 xv = ((const float4*)(x + (size_t)r * N_FEAT))[lane];  // 32 lanes * 16B = 512B row
    float* dst = agg + (size_t)c * N_FEAT + lane * 4;
    atomicAdd(dst + 0, xv.x * w);
    atomicAdd(dst + 1, xv.y * w);
    atomicAdd(dst + 2, xv.z * w);
    atomicAdd(dst + 3, xv.w * w);
}

// self-loop init: agg[i] = x[i] * dis[i]^2
__global__ void k_sg_init(const float* __restrict__ x, const float* __restrict__ dis,
                          float* __restrict__ agg, int n) {
    int i = blockIdx.x * blockDim.x + threadIdx.x;
    if (i >= n * N_FEAT) return;
    float d = dis[i >> 7];
    agg[i] = x[i] * d * d;
}

// agg[c] += x[r] * dis[r]*dis[c]
__global__ void k_sg_scatter(const float* __restrict__ x, const int* __restrict__ row,
                             const int* __restrict__ col, const float* __restrict__ dis,
                             float* __restrict__ agg, int nE) {
    int e = (blockIdx.x * blockDim.x + threadIdx.x) >> 5;
    if (e >= nE) return;
    int lane = threadIdx.x & 31;
    int r = row[e], c = col[e];
    float w = dis[r] * dis[c];
    float4 xv = ((const float4*)(x + (size_t)r * N_FEAT))[lane];
    float* dst = agg + (size_t)c * N_FEAT + lane * 4;
    atomicAdd(dst + 0, xv.x * w);
    atomicAdd(dst + 1, xv.y * w);
    atomicAdd(dst + 2, xv.z * w);
    atomicAdd(dst + 3, xv.w * w);
}

// ---------------- dense GEMM via V_WMMA_F32_16X16X4_F32 ----------------
// One pass: stage 80x128 A tile into LDS via async-to-LDS B128 copies
// (per-lane LDS addresses keep the 16B/row padding), then accumulate
// c[mt] += A_tile @ W^T for 5 M-tiles sharing each B fragment.
__device__ __forceinline__ void gemm_pass(const float* __restrict__ A,
                                          const float* __restrict__ W,
                                          float* __restrict__ aLds,
                                          int mBase, int tid, v8f c[M_TILES]) {
    // Async-stage A tile (ROWS_PER_BLOCK x 128, row-padded to A_STRIDE floats).
    for (int i = tid; i < ROWS_PER_BLOCK * (N_FEAT / 4); i += 256) {
        int m = i >> 5;                 // row 0..79
        int v = i & 31;                 // float4 slot within row
        const float* g = A + (size_t)(mBase + m) * N_FEAT + (v << 2);
        unsigned laddr = (unsigned)(uintptr_t)(aLds + m * A_STRIDE + (v << 2));
        asm volatile("global_load_async_to_lds_b128 %0, %1, off"
                     :: "v"(laddr), "v"(g) : "memory");
    }
    asm volatile("s_wait_asynccnt 0x0" ::: "memory");
    __syncthreads();

    const int lane = tid & 31;
    const int half = lane >> 4;         // lanes 16-31 carry K+2 / M+8
    const int l16  = lane & 15;
    const int ncol = (tid >> 5) * 16 + l16;   // wave -> 16-col tile

    for (int k = 0; k < N_FEAT; k += 4) {
        int k0 = k + half * 2;          // even -> 8B-aligned loads
        float2 wv = *(const float2*)(W + (size_t)ncol * N_FEAT + k0);  // B[k][n]=W[n][k]
        v2f b;
        b.x = wv.x;
        b.y = wv.y;
        #pragma unroll
        for (int mt = 0; mt < M_TILES; ++mt) {
            float2 av = *(const float2*)(aLds + (size_t)(mt * 16 + l16) * A_STRIDE + k0);
            v2f a;
            a.x = av.x;                 // A frag: M=l16, K=k0,k0+1
            a.y = av.y;
            c[mt] = __builtin_amdgcn_wmma_f32_16x16x4_f32(false, a, false, b, (short)0,
                                                          c[mt], false, false);
        }
    }
}

// out[m][n] = sum_k A0[m][k]*W0[n][k] (+ A1[m][k]*W1[n][k]) (+ bias[n])
// Block = 256 threads (8 waves); block owns an 80-row stripe; wave w owns
// cols [16w,16w+16). DUAL runs two passes over one LDS tile buffer,
// accumulating into the same C registers. 50000 = 625*80 -> no partial tiles.
template <bool DUAL>
__global__ void wmma_gemm(const float* __restrict__ A0, const float* __restrict__ W0,
                          const float* __restrict__ A1, const float* __restrict__ W1,
                          const float* __restrict__ bias, float* __restrict__ out) {
    __shared__ float aLds[ROWS_PER_BLOCK * A_STRIDE];     // 42,240 B
    const int tid   = threadIdx.x;
    const int mBase = blockIdx.x * ROWS_PER_BLOCK;

    __builtin_prefetch(W0, 0, 1);                          // global_prefetch_b8
    if (DUAL) __builtin_prefetch(W1, 0, 1);

    v8f c[M_TILES] = {};
    gemm_pass(A0, W0, aLds, mBase, tid, c);
    if (DUAL) {
        __syncthreads();                                   // all waves done with pass-0 tile
        gemm_pass(A1, W1, aLds, mBase, tid, c);
    }

    const int lane = tid & 31;
    const int half = lane >> 4;
    const int l16  = lane & 15;
    const int ncol = (tid >> 5) * 16 + l16;
    float bv = bias ? bias[ncol] : 0.0f;
    // C/D layout: VGPR j -> row M=j (lanes 0-15) / M=j+8 (lanes 16-31), col=lane&15
    #pragma unroll
    for (int mt = 0; mt < M_TILES; ++mt) {
        #pragma unroll
        for (int vg = 0; vg < 8; ++vg) {
            int rowi = mBase + mt * 16 + vg + half * 8;
            out[(size_t)rowi * N_FEAT + ncol] = c[mt][vg] + bv;
        }
    }
}

// ---------------- BatchNorm (training stats) + ReLU ----------------
__global__ void k_bnstats(const float* __restrict__ h, float* __restrict__ sum,
                          float* __restrict__ ssq, int n) {
    int col = threadIdx.x;                         // 128 threads = 4 waves
    float s = 0.0f, s2 = 0.0f;
    for (int r = blockIdx.x; r < n; r += gridDim.x) {
        float v = h[(size_t)r * N_FEAT + col];     // coalesced row reads
        s  += v;
        s2 += v * v;
    }
    atomicAdd(&sum[col], s);
    atomicAdd(&ssq[col], s2);
}

__global__ void k_bnrelu(float* __restrict__ h, const float* __restrict__ sum,
                         const float* __restrict__ ssq, const float* __restrict__ gamma,
                         const float* __restrict__ beta, int n) {
    int i = blockIdx.x * blockDim.x + threadIdx.x;
    if (i >= n * N_FEAT) return;
    int cc = i & 127;
    float invN = 1.0f / (float)n;
    float mu   = sum[cc] * invN;
    float var  = fmaxf(ssq[cc] * invN - mu * mu, 0.0f);
    float rs   = rsqrtf(var + BN_EPS);
    float v    = (h[i] - mu) * rs * gamma[cc] + beta[cc];
    h[i] = fmaxf(v, 0.0f);
}

// ---------------------------------------------------------------------------
extern "C" void kernel_launch(void* const* d_in, const int* in_sizes, int n_in,
                              void* d_out, int out_size, void* d_ws, size_t ws_size,
                              hipStream_t stream) {
    const int*   x_idx  = (const int*)d_in[0];
    const int*   edge   = (const int*)d_in[1];
    const float* emb    = (const float*)d_in[2];
    const float* W_out  = (const float*)d_in[3];
    const float* W_root = (const float*)d_in[4];
    const float* bn0_g  = (const float*)d_in[5];
    const float* bn0_b  = (const float*)d_in[6];
    const float* W_sg1  = (const float*)d_in[7];
    const float* b_sg1  = (const float*)d_in[8];
    const float* bn1_g  = (const float*)d_in[9];
    const float* bn1_b  = (const float*)d_in[10];
    const float* W_sg2  = (const float*)d_in[11];
    const float* b_sg2  = (const float*)d_in[12];

    const int E   = in_sizes[1] / 2;
    const int N   = in_sizes[2] / N_FEAT;          // 50000 (multiple of 80)
    const int* row = edge;
    const int* col = edge + E;

    // workspace layout (floats): deg | dis | dinv | sum(128) | ssq(128) | x | agg | h
    float* deg  = (float*)d_ws;
    float* dis  = deg  + N;
    float* dinv = dis  + N;
    float* sum  = dinv + N;
    float* ssq  = sum  + 128;
    float* xbuf = ssq  + 128;
    float* agg  = xbuf + (size_t)N * N_FEAT;
    float* hbuf = agg  + (size_t)N * N_FEAT;

    const int NF     = N * N_FEAT;                 // 6.4M
    const int gNF    = (NF + 255) / 256;
    const int gN     = (N + 255) / 256;
    const int gE     = (E + 255) / 256;
    const int gEwave = (E + 7) / 8;                // 1 wave/edge, 8 edges/block
    const int gTile  = N / ROWS_PER_BLOCK;         // 625 row stripes

    // prep: gather x, degrees, norms
    k_gather<<<(N * 32 + 255) / 256, 256, 0, stream>>>(emb, x_idx, xbuf, N);
    k_zero<<<gN, 256, 0, stream>>>(deg, N);
    k_deg<<<gE, 256, 0, stream>>>(col, deg, E);
    k_prep<<<gN, 256, 0, stream>>>(deg, dinv, dis, N);

    // layer 0: ClusterGCN -> BN -> ReLU
    k_zero<<<gNF, 256, 0, stream>>>(agg, NF);
    k_cluster_scatter<<<gEwave, 256, 0, stream>>>(xbuf, row, col, dinv, agg, E);
    wmma_gemm<true><<<gTile, 256, 0, stream>>>(agg, W_out, xbuf, W_root, nullptr, hbuf);
    k_zero<<<1, 256, 0, stream>>>(sum, 256);       // zero sum+ssq (contiguous)
    k_bnstats<<<512, 128, 0, stream>>>(hbuf, sum, ssq, N);
    k_bnrelu<<<gNF, 256, 0, stream>>>(hbuf, sum, ssq, bn0_g, bn0_b, N);

    // layer 1: SGConv -> BN -> ReLU
    k_sg_init<<<gNF, 256, 0, stream>>>(hbuf, dis, agg, N);
    k_sg_scatter<<<gEwave, 256, 0, stream>>>(hbuf, row, col, dis, agg, E);
    wmma_gemm<false><<<gTile, 256, 0, stream>>>(agg, W_sg1, nullptr, nullptr, b_sg1, hbuf);
    k_zero<<<1, 256, 0, stream>>>(sum, 256);
    k_bnstats<<<512, 128, 0, stream>>>(hbuf, sum, ssq, N);
    k_bnrelu<<<gNF, 256, 0, stream>>>(hbuf, sum, ssq, bn1_g, bn1_b, N);

    // layer 2: final SGConv -> d_out
    k_sg_init<<<gNF, 256, 0, stream>>>(hbuf, dis, agg, N);
    k_sg_scatter<<<gEwave, 256, 0, stream>>>(hbuf, row, col, dis, agg, E);
    wmma_gemm<false><<<gTile, 256, 0, stream>>>(agg, W_sg2, nullptr, nullptr, b_sg2,
                                                (float*)d_out);
}